// StochasticPool2d_57475252355389
// MI455X (gfx1250) — compile-verified
//
#include <hip/hip_runtime.h>

// StochasticPool2d, memory-bound streaming kernel:
//   x:      [32, 64, 224, 224] f32   (411 MB, read once)
//   gumbel: [32, 64, 112, 112, 4] f32 (411 MB, read once)
//   out:    [32, 64, 112, 112] f32   (103 MB, written once)
// ~925 MB @ 23.3 TB/s => ~40 us floor. No matrix structure -> no WMMA.
// 8 outputs/thread; all global traffic is 128-bit non-temporal; VALU math
// minimized (log folded into FMA, single cndmask per element, pairwise argmax).

typedef float v4f __attribute__((ext_vector_type(4)));

namespace {
constexpr unsigned BC  = 32u * 64u;   // fused batch*channel
constexpr unsigned H   = 224u;
constexpr unsigned W   = 224u;
constexpr unsigned NH  = 112u;
constexpr unsigned NW  = 112u;
constexpr unsigned J8  = NW / 8u;                 // 14 groups of 8 outputs per row
constexpr unsigned TOTAL_THREADS = BC * NH * J8;  // 3,211,264
constexpr unsigned BLOCK = 256u;                  // 8 waves (wave32)
}

// One 2x2 window: score_k = ln(where(v_k>0, v_k, 1)) + g_k, return v[argmax].
// ln(1)=0 => s_k = v_k>0 ? fma(log2(v_k), ln2, g_k) : g_k  (log runs
// speculatively; NaN/-inf from v<=0 is discarded by the select).
// First-index tie-break preserved with strict '>' everywhere.
__device__ __forceinline__ float pick4(float v0, float v1, float v2, float v3,
                                       v4f g) {
    constexpr float LN2 = 0.69314718055994530942f;
    const float s0 = (v0 > 0.0f) ? __builtin_fmaf(__builtin_amdgcn_logf(v0), LN2, g.x) : g.x;
    const float s1 = (v1 > 0.0f) ? __builtin_fmaf(__builtin_amdgcn_logf(v1), LN2, g.y) : g.y;
    const float s2 = (v2 > 0.0f) ? __builtin_fmaf(__builtin_amdgcn_logf(v2), LN2, g.z) : g.z;
    const float s3 = (v3 > 0.0f) ? __builtin_fmaf(__builtin_amdgcn_logf(v3), LN2, g.w) : g.w;

    const float b01 = __builtin_fmaxf(s0, s1);
    const float v01 = (s1 > s0) ? v1 : v0;       // tie -> earlier index
    const float b23 = __builtin_fmaxf(s2, s3);
    const float v23 = (s3 > s2) ? v3 : v2;
    return (b23 > b01) ? v23 : v01;              // tie -> earlier pair
}

__global__ __launch_bounds__(256)
void stochpool2d_nt_kernel(const float* __restrict__ x,
                           const float* __restrict__ gum,
                           float* __restrict__ out)
{
    const unsigned tid = blockIdx.x * BLOCK + threadIdx.x;
    if (tid >= TOTAL_THREADS) return;

    // tid -> (bc, output row i, group-of-8-columns j8); 32-bit const-div math.
    const unsigned j8 = tid % J8;
    const unsigned t  = tid / J8;
    const unsigned i  = t % NH;
    const unsigned bc = t / NH;

    const size_t xRow0 = (size_t)(bc * H + 2u * i) * W + 16u * j8;      // 64B aligned
    const size_t gBase = ((size_t)(bc * NH + i) * NW + 8u * j8) * 4u;   // 128B aligned
    const size_t oBase = (size_t)(bc * NH + i) * NW + 8u * j8;          // 32B aligned

    const v4f* xr0 = (const v4f*)(x + xRow0);        // row 2i   : 16 floats
    const v4f* xr1 = (const v4f*)(x + xRow0 + W);    // row 2i+1 : 16 floats
    const v4f* gp  = (const v4f*)(gum + gBase);      // 32 gumbels
    v4f*       op  = (v4f*)(out + oBase);            // 8 outputs

    // 16 x b128 non-temporal loads (single-use stream: keep it out of L2/WGP$).
    const v4f a0 = __builtin_nontemporal_load(xr0 + 0);
    const v4f a1 = __builtin_nontemporal_load(xr0 + 1);
    const v4f a2 = __builtin_nontemporal_load(xr0 + 2);
    const v4f a3 = __builtin_nontemporal_load(xr0 + 3);
    const v4f b0 = __builtin_nontemporal_load(xr1 + 0);
    const v4f b1 = __builtin_nontemporal_load(xr1 + 1);
    const v4f b2 = __builtin_nontemporal_load(xr1 + 2);
    const v4f b3 = __builtin_nontemporal_load(xr1 + 3);
    const v4f g0 = __builtin_nontemporal_load(gp + 0);
    const v4f g1 = __builtin_nontemporal_load(gp + 1);
    const v4f g2 = __builtin_nontemporal_load(gp + 2);
    const v4f g3 = __builtin_nontemporal_load(gp + 3);
    const v4f g4 = __builtin_nontemporal_load(gp + 4);
    const v4f g5 = __builtin_nontemporal_load(gp + 5);
    const v4f g6 = __builtin_nontemporal_load(gp + 6);
    const v4f g7 = __builtin_nontemporal_load(gp + 7);

    // Window k uses row0[2k,2k+1], row1[2k,2k+1], gumbel[4k..4k+3].
    v4f r0, r1;
    r0.x = pick4(a0.x, a0.y, b0.x, b0.y, g0);
    r0.y = pick4(a0.z, a0.w, b0.z, b0.w, g1);
    r0.z = pick4(a1.x, a1.y, b1.x, b1.y, g2);
    r0.w = pick4(a1.z, a1.w, b1.z, b1.w, g3);
    r1.x = pick4(a2.x, a2.y, b2.x, b2.y, g4);
    r1.y = pick4(a2.z, a2.w, b2.z, b2.w, g5);
    r1.z = pick4(a3.x, a3.y, b3.x, b3.y, g6);
    r1.w = pick4(a3.z, a3.w, b3.z, b3.w, g7);

    // 2 x b128 non-temporal stores.
    __builtin_nontemporal_store(r0, op + 0);
    __builtin_nontemporal_store(r1, op + 1);
}

extern "C" void kernel_launch(void* const* d_in, const int* in_sizes, int n_in,
                              void* d_out, int out_size, void* d_ws, size_t ws_size,
                              hipStream_t stream) {
    (void)in_sizes; (void)n_in; (void)out_size; (void)d_ws; (void)ws_size;
    const float* x   = (const float*)d_in[0];
    const float* gum = (const float*)d_in[1];
    float*       out = (float*)d_out;

    const unsigned grid = (TOTAL_THREADS + BLOCK - 1u) / BLOCK;  // 12544 blocks
    hipLaunchKernelGGL(stochpool2d_nt_kernel, dim3(grid), dim3(BLOCK), 0, stream,
                       x, gum, out);
}